// SubmanifoldSparseRPN_4337916970098
// MI455X (gfx1250) — compile-verified
//
#include <hip/hip_runtime.h>

// ---------------------------------------------------------------------------
// SubmanifoldSparseRPN on MI455X (gfx1250, wave32):
//   6x [ SubMConv3x3 (implicit GEMM via V_WMMA_F32_16X16X4_F32, fp32)
//        -> masked BN (global masked stats) -> ReLU -> re-mask ]
// Conv is structured as 9 per-tap GEMM passes over K=Ci so the hot loop is
// just { 2 B loads, 1 float2 A load per Co-tile, WMMA } with no index decode.
// fp32 WMMA keeps reference precision; working set is L2-resident (192MB L2).
// ---------------------------------------------------------------------------

typedef float v2f __attribute__((ext_vector_type(2)));
typedef float v8f __attribute__((ext_vector_type(8)));

#define NB 4
#define HH 192
#define WW 192
#define HWSZ (HH * WW)
#define WT 12          // WW / 16 position tiles per row
#define BN_EPS 1e-3f

// ---------------------------------------------------------------------------
// Weight repack: w[co][ci][kh][kw]  ->  wtr[(tap*Co + co)*Ci + ci]
// so per-tap A fragments are contiguous float2 along ci.
// ---------------------------------------------------------------------------
__global__ void repack_w_kernel(const float* __restrict__ w,
                                float* __restrict__ wtr, int Ci, int Co)
{
    const int i = blockIdx.x * blockDim.x + threadIdx.x;
    const int total = Co * Ci * 9;
    if (i < total) {
        const int co  = i / (Ci * 9);
        const int r   = i - co * (Ci * 9);
        const int ci  = r / 9;
        const int tap = r - ci * 9;
        wtr[(tap * Co + co) * Ci + ci] = w[i];
    }
}

// ---------------------------------------------------------------------------
// Implicit-GEMM 3x3 conv, one wave per 16-position tile, all Co channels.
// Accumulate over 9 taps; per tap, GEMM over K=CI in chunks of 4 with
// V_WMMA_F32_16X16X4_F32. Out-of-range rows are skipped with a wave-uniform
// branch (EXEC stays all-ones); out-of-range columns are zeroed per lane.
// Output is multiplied by the active-site mask on store.
// ---------------------------------------------------------------------------
template <int CI, int CO>
__global__ __launch_bounds__(32) void conv_wmma_kernel(
    const float* __restrict__ in, float* __restrict__ out,
    const float* __restrict__ wtr, const float* __restrict__ mask)
{
    constexpr int NCO16 = CO / 16;

    const int lane = threadIdx.x;
    const int half = lane >> 4;   // 0: K rows {0,1}, 1: K rows {2,3}
    const int lcol = lane & 15;   // matrix column / output-channel row

    int t = blockIdx.x;
    const int wt = t % WT;  t /= WT;
    const int h  = t % HH;
    const int n  = t / HH;
    const int w0 = wt << 4;

    const v8f vzero = {};
    v8f acc[NCO16];
#pragma unroll
    for (int cb = 0; cb < NCO16; ++cb) acc[cb] = vzero;

    const float* inN = in + n * CI * HWSZ;

    for (int tap = 0; tap < 9; ++tap) {
        const int kh = tap / 3;
        const int kw = tap - 3 * kh;
        const int ih = h + kh - 1;                 // wave-uniform
        if ((unsigned)ih >= (unsigned)HH) continue; // uniform skip, EXEC intact

        const int  iw  = w0 + lcol + kw - 1;       // per-lane column
        const bool ok  = (unsigned)iw < (unsigned)WW;
        const float okf = ok ? 1.0f : 0.0f;

        // B pointer for this lane: rows ci = c0 + 2*half (+1), col iw.
        const float* bptr = inN + (half << 1) * HWSZ + ih * WW + (ok ? iw : 0);
        // A pointer for this lane: wtr[(tap*CO + lcol)*CI + 2*half], cb adds 16*CI.
        const float* aptr = wtr + (tap * CO + lcol) * CI + (half << 1);

        for (int c0 = 0; c0 < CI; c0 += 4) {
            v2f bfrag;
            bfrag[0] = bptr[c0 * HWSZ] * okf;
            bfrag[1] = bptr[(c0 + 1) * HWSZ] * okf;
#pragma unroll
            for (int cb = 0; cb < NCO16; ++cb) {
                const float2 aw = *(const float2*)(aptr + cb * 16 * CI + c0);
                v2f afrag; afrag[0] = aw.x; afrag[1] = aw.y;
                acc[cb] = __builtin_amdgcn_wmma_f32_16x16x4_f32(
                    false, afrag, false, bfrag, (short)0, acc[cb], false, false);
            }
        }
    }

    // D layout: VGPR v, lanes 0-15 -> (M=v, N=lane); lanes 16-31 -> (M=v+8).
    const float mval = mask[(n * HH + h) * WW + w0 + lcol];
#pragma unroll
    for (int cb = 0; cb < NCO16; ++cb) {
#pragma unroll
        for (int v = 0; v < 8; ++v) {
            const int co = (cb << 4) + (half << 3) + v;
            out[((n * CO + co) * HH + h) * WW + w0 + lcol] = acc[cb][v] * mval;
        }
    }
}

// ---------------------------------------------------------------------------
// Active-site mask from input x (C=32) + count of active sites.
// ---------------------------------------------------------------------------
__global__ void mask_count_kernel(const float* __restrict__ x,
                                  float* __restrict__ mask,
                                  float* __restrict__ cnt)
{
    __shared__ float sm[256];
    const int i = blockIdx.x * blockDim.x + threadIdx.x;
    float m = 0.0f;
    if (i < NB * HWSZ) {
        const int n = i / HWSZ, hw = i - n * HWSZ;
        const float* p = x + (size_t)n * 32 * HWSZ + hw;
        bool any = false;
#pragma unroll
        for (int c = 0; c < 32; ++c) any |= (p[c * HWSZ] != 0.0f);
        m = any ? 1.0f : 0.0f;
        mask[i] = m;
    }
    sm[threadIdx.x] = m;
    __syncthreads();
    for (int s = 128; s > 0; s >>= 1) {
        if (threadIdx.x < s) sm[threadIdx.x] += sm[threadIdx.x + s];
        __syncthreads();
    }
    if (threadIdx.x == 0) atomicAdd(cnt, sm[0]);
}

__global__ void zero_kernel(float* __restrict__ p, int n)
{
    const int i = blockIdx.x * blockDim.x + threadIdx.x;
    if (i < n) p[i] = 0.0f;
}

// ---------------------------------------------------------------------------
// Per-channel sum / sumsq over masked conv output (inactive sites are 0,
// so plain sums over everything equal sums over active sites).
// One block per (n, c) plane.
// ---------------------------------------------------------------------------
__global__ void stats_kernel(const float* __restrict__ y,
                             float* __restrict__ sum,
                             float* __restrict__ sumsq, int Co)
{
    __shared__ float s1[256], s2[256];
    const int bid = blockIdx.x;            // n*Co + c
    const int c = bid % Co;
    const float* plane = y + (size_t)bid * HWSZ;
    float a = 0.0f, b = 0.0f;
    for (int i = threadIdx.x; i < HWSZ; i += 256) {
        const float v = plane[i];
        a += v; b += v * v;
    }
    s1[threadIdx.x] = a; s2[threadIdx.x] = b;
    __syncthreads();
    for (int s = 128; s > 0; s >>= 1) {
        if (threadIdx.x < s) {
            s1[threadIdx.x] += s1[threadIdx.x + s];
            s2[threadIdx.x] += s2[threadIdx.x + s];
        }
        __syncthreads();
    }
    if (threadIdx.x == 0) {
        atomicAdd(&sum[c],   s1[0]);
        atomicAdd(&sumsq[c], s2[0]);
    }
}

__global__ void finalize_kernel(const float* __restrict__ sum,
                                const float* __restrict__ sumsq,
                                const float* __restrict__ cnt,
                                const float* __restrict__ gamma,
                                const float* __restrict__ beta,
                                float* __restrict__ scale,
                                float* __restrict__ shift, int Co)
{
    const int c = threadIdx.x;
    if (c < Co) {
        const float n    = cnt[0];
        const float mean = sum[c] / n;
        const float var  = sumsq[c] / n - mean * mean;   // biased variance
        const float sc   = gamma[c] * rsqrtf(var + BN_EPS);
        scale[c] = sc;
        shift[c] = beta[c] - mean * sc;
    }
}

__global__ void bnrelu_kernel(float* __restrict__ y,
                              const float* __restrict__ mask,
                              const float* __restrict__ scale,
                              const float* __restrict__ shift,
                              int Co, int total)
{
    const int i = blockIdx.x * blockDim.x + threadIdx.x;
    if (i >= total) return;
    const int hw = i % HWSZ;
    const int nc = i / HWSZ;
    const int c = nc % Co;
    const int n = nc / Co;
    const float m = mask[n * HWSZ + hw];
    const float v = fmaf(y[i], scale[c], shift[c]);
    y[i] = fmaxf(v, 0.0f) * m;
}

// ---------------------------------------------------------------------------
// Host side: one layer = repack -> conv(WMMA) -> stats -> finalize -> bn+relu
// ---------------------------------------------------------------------------
static void run_layer(const float* in, float* out, const float* w,
                      const float* gamma, const float* beta, int Ci, int Co,
                      const float* mask, const float* cnt, float* wtr,
                      float* sum, float* sumsq, float* scale, float* shift,
                      hipStream_t stream)
{
    const int wTotal = Co * Ci * 9;
    repack_w_kernel<<<(wTotal + 255) / 256, 256, 0, stream>>>(w, wtr, Ci, Co);

    const int posTiles = NB * HH * WT;     // 9216 tiles of 16 positions
    if (Ci == 32 && Co == 32)
        conv_wmma_kernel<32, 32><<<posTiles, 32, 0, stream>>>(in, out, wtr, mask);
    else if (Ci == 32 && Co == 64)
        conv_wmma_kernel<32, 64><<<posTiles, 32, 0, stream>>>(in, out, wtr, mask);
    else
        conv_wmma_kernel<64, 64><<<posTiles, 32, 0, stream>>>(in, out, wtr, mask);

    zero_kernel<<<1, 128, 0, stream>>>(sum, 128);  // sum[64] + sumsq[64] contiguous
    stats_kernel<<<NB * Co, 256, 0, stream>>>(out, sum, sumsq, Co);
    finalize_kernel<<<1, 64, 0, stream>>>(sum, sumsq, cnt, gamma, beta,
                                          scale, shift, Co);
    const int total = NB * Co * HWSZ;
    bnrelu_kernel<<<(total + 255) / 256, 256, 0, stream>>>(out, mask, scale,
                                                           shift, Co, total);
}

extern "C" void kernel_launch(void* const* d_in, const int* in_sizes, int n_in,
                              void* d_out, int out_size, void* d_ws, size_t ws_size,
                              hipStream_t stream)
{
    (void)in_sizes; (void)n_in; (void)out_size; (void)ws_size;

    // setup_inputs order: x, weights[0..5], gammas[0..5], betas[0..5]
    const float* x = (const float*)d_in[0];
    const float* wts[6]; const float* gms[6]; const float* bts[6];
    for (int i = 0; i < 6; ++i) {
        wts[i] = (const float*)d_in[1 + i];
        gms[i] = (const float*)d_in[7 + i];
        bts[i] = (const float*)d_in[13 + i];
    }
    float* out = (float*)d_out;

    // Workspace: act(4*64*HW) | mask(4*HW) | cnt(64 pad) | sum(64) | sumsq(64)
    //            | scale(64) | shift(64) | wtr(9*64*64)  ~= 38.6 MB
    float* ws    = (float*)d_ws;
    float* act   = ws;
    float* mask  = act  + (size_t)NB * 64 * HWSZ;
    float* cnt   = mask + (size_t)NB * HWSZ;
    float* sum   = cnt  + 64;
    float* sumsq = sum  + 64;
    float* scale = sumsq + 64;
    float* shift = scale + 64;
    float* wtr   = shift + 64;

    zero_kernel<<<1, 32, 0, stream>>>(cnt, 1);
    mask_count_kernel<<<(NB * HWSZ) / 256, 256, 0, stream>>>(x, mask, cnt);

    // Ping-pong act (ws) <-> d_out; layer 6 lands in d_out.
    run_layer(x,   act, wts[0], gms[0], bts[0], 32, 32, mask, cnt, wtr, sum, sumsq, scale, shift, stream);
    run_layer(act, out, wts[1], gms[1], bts[1], 32, 32, mask, cnt, wtr, sum, sumsq, scale, shift, stream);
    run_layer(out, act, wts[2], gms[2], bts[2], 32, 64, mask, cnt, wtr, sum, sumsq, scale, shift, stream);
    run_layer(act, out, wts[3], gms[3], bts[3], 64, 64, mask, cnt, wtr, sum, sumsq, scale, shift, stream);
    run_layer(out, act, wts[4], gms[4], bts[4], 64, 64, mask, cnt, wtr, sum, sumsq, scale, shift, stream);
    run_layer(act, out, wts[5], gms[5], bts[5], 64, 64, mask, cnt, wtr, sum, sumsq, scale, shift, stream);
}